// SSIMLoss_47004122088007
// MI455X (gfx1250) — compile-verified
//
#include <hip/hip_runtime.h>

// SSIM loss, B=32,C=1,640x640 f32, 7x7 box window, VALID -> 634x634.
// Memory-bound (~105 MB @ 23.3 TB/s ~= 4.5us floor). Per 64x64-output tile:
//   Phase A: TDM tensor_load_to_lds DMAs the 70x70 x/y tiles into LDS
//            (tensor_dim clamped to the image remainder -> HW zero-fill OOB).
//   Phase B: vertical 7-row sliding sums of (x,y,xx,yy,xy) on VALU.
//   Phase C: horizontal 7-tap sum as a Toeplitz-band matmul on the matrix
//            units: 6 chained V_WMMA_F32_16X16X4_F32 per 16x16 block, then
//            SSIM pointwise directly on the in-register accumulators.

typedef __attribute__((ext_vector_type(2))) float v2f;
typedef __attribute__((ext_vector_type(8))) float v8f;
typedef __attribute__((ext_vector_type(4))) unsigned int v4u;
typedef __attribute__((ext_vector_type(4))) int v4i;
typedef __attribute__((ext_vector_type(8))) int v8i;

#define IMG   640
#define WOUT  634          // 640 - 7 + 1
#define TILE  64           // outputs per tile edge
#define NT    10           // ceil(634/64)
#define IN_T  70           // TILE + 6 input extent
#define VST   70           // LDS row stride (70 % 64 = 6 -> conflict-free column walks)
#define NBLK  256

// LDS float layout: sx[70*70] | sy[70*70] | V[5][64*70] | pad[8] | red[8]
#define OFF_SY   (IN_T * IN_T)                 // 4900
#define OFF_V    (2 * IN_T * IN_T)             // 9800
#define VQ       (TILE * VST)                  // 4480 per quantity
#define OFF_RED  (OFF_V + 5 * VQ + 8)          // 32208
#define LDS_FLTS (OFF_RED + 8)                 // 32216 -> 128864 bytes (<320KB/WG)

// Issue one TDM 2D tile load: 70x70 f32 tile at gptr (row stride IMG floats)
// into LDS byte offset lds_off. rem_x/rem_y = remaining in-image extent from
// the tile origin; TDM returns zero for reads beyond tensor_dim (HW zero-fill)
// so edge tiles never touch out-of-bounds memory. D# per CDNA5 ISA.
__device__ __forceinline__ void tdm_load_tile_2d(unsigned int lds_off,
                                                 const float* gptr,
                                                 unsigned int rem_x,
                                                 unsigned int rem_y)
{
    const unsigned long long ga = (unsigned long long)(uintptr_t)gptr;
    // Group 0: count=1 | lds_addr | global_addr[56:0] | type=2 ("image")
    v4u g0;
    g0[0] = 1u;
    g0[1] = lds_off;
    g0[2] = (unsigned int)ga;
    g0[3] = ((unsigned int)(ga >> 32) & 0x01FFFFFFu) | 0x80000000u;
    // Group 1: wg_mask=0, data_size=2 (4B), no pad/iterate/barrier,
    // tensor_dim0/1 = remaining extent, tile 70x70, dim0 stride = 640.
    v8i g1;
    g1[0] = (int)(2u << 16);                                    // data_size=4B
    g1[1] = (int)((rem_x & 0xFFFFu) << 16);                     // dim0 lo16
    g1[2] = (int)((rem_x >> 16) | ((rem_y & 0xFFFFu) << 16));   // dim0 hi / dim1 lo
    g1[3] = (int)((rem_y >> 16) | ((unsigned)IN_T << 16));      // dim1 hi / tile_dim0
    g1[4] = (int)IN_T;                                          // tile_dim1 (tile_dim2=0)
    g1[5] = (int)IMG;                                           // dim0_stride lo32
    g1[6] = 0;                                                  // dim0_stride hi / dim1_stride lo
    g1[7] = 0;
    const v4i z4 = {0, 0, 0, 0};
#if __clang_major__ >= 23
    const v8i z8 = {0, 0, 0, 0, 0, 0, 0, 0};
    __builtin_amdgcn_tensor_load_to_lds(g0, g1, z4, z4, z8, 0);
#else
    __builtin_amdgcn_tensor_load_to_lds(g0, g1, z4, z4, 0);
#endif
}

__global__ __launch_bounds__(NBLK)
void ssim_tile_kernel(const float* __restrict__ x, const float* __restrict__ y,
                      const float* __restrict__ dr, float* __restrict__ partial)
{
    extern __shared__ float lds[];
    float* sx = lds;
    float* sy = lds + OFF_SY;
    float* V  = lds + OFF_V;
    float* red = lds + OFF_RED;

    const int tid  = threadIdx.x;
    const int lane = tid & 31;           // wave32
    const int wid  = tid >> 5;           // 8 waves
    const int tileId = blockIdx.x;       // 0..99
    const int b      = blockIdx.y;       // batch
    const int tx = tileId % NT, ty = tileId / NT;
    const int gx0 = tx * TILE, gy0 = ty * TILE;

    const float* __restrict__ xb = x + (size_t)b * IMG * IMG;
    const float* __restrict__ yb = y + (size_t)b * IMG * IMG;

    // ---- Phase A: TDM DMA of the two 70x70 input tiles into LDS.
    // Wave 0 moves x, wave 1 moves y (parallel DMA engines); each waits on
    // its own TENSORcnt, then the barrier publishes LDS to the workgroup. ----
    const unsigned int lds_base = (unsigned int)(uintptr_t)lds;  // LDS byte offset
    if (wid == 0) {
        tdm_load_tile_2d(lds_base,
                         xb + (size_t)gy0 * IMG + gx0,
                         (unsigned)(IMG - gx0), (unsigned)(IMG - gy0));
        __builtin_amdgcn_s_wait_tensorcnt(0);
    } else if (wid == 1) {
        tdm_load_tile_2d(lds_base + OFF_SY * (unsigned)sizeof(float),
                         yb + (size_t)gy0 * IMG + gx0,
                         (unsigned)(IMG - gx0), (unsigned)(IMG - gy0));
        __builtin_amdgcn_s_wait_tensorcnt(0);
    }
    __syncthreads();

    // ---- Phase B: vertical 7-row sliding sums of (x, y, xx, yy, xy).
    // 280 units = (column c, 16-row quarter); all 256 threads active. ----
    for (int u = tid; u < 4 * IN_T; u += NBLK) {
        const int c  = u % IN_T;
        const int r0 = (u / IN_T) * (TILE / 4);     // 0,16,32,48
        float s1 = 0.f, s2 = 0.f, s3 = 0.f, s4 = 0.f, s5 = 0.f;
        #pragma unroll
        for (int k = 0; k < 7; ++k) {
            const float a = sx[(r0 + k) * VST + c];
            const float g = sy[(r0 + k) * VST + c];
            s1 += a; s2 += g; s3 += a * a; s4 += g * g; s5 += a * g;
        }
        int o = r0 * VST + c;
        V[0 * VQ + o] = s1; V[1 * VQ + o] = s2; V[2 * VQ + o] = s3;
        V[3 * VQ + o] = s4; V[4 * VQ + o] = s5;
        for (int r = r0 + 1; r < r0 + TILE / 4; ++r) {
            const float ao = sx[(r - 1) * VST + c], go = sy[(r - 1) * VST + c];
            const float an = sx[(r + 6) * VST + c], gn = sy[(r + 6) * VST + c];
            s1 += an - ao;
            s2 += gn - go;
            s3 += an * an - ao * ao;
            s4 += gn * gn - go * go;
            s5 += an * gn - ao * go;
            o = r * VST + c;
            V[0 * VQ + o] = s1; V[1 * VQ + o] = s2; V[2 * VQ + o] = s3;
            V[3 * VQ + o] = s4; V[4 * VQ + o] = s5;
        }
    }
    __syncthreads();

    // ---- Phase C: horizontal 7-tap sum as D = A(16x4-slices) x T(4x16) via
    // chained V_WMMA_F32_16X16X4_F32; T = 1/49-scaled Toeplitz band. ----
    const int khalf = lane >> 4;         // A: K={0,1} vs {2,3}; D rows 0..7 vs 8..15
    const int ln    = lane & 15;

    const float drb = dr[b];
    const float c1 = (0.01f * drb) * (0.01f * drb);
    const float c2 = (0.03f * drb) * (0.03f * drb);
    const float CN = 49.0f / 48.0f;      // WIN^2/(WIN^2-1)

    // Constant B chunks: B 4x16 layout -> VGPR v holds row K = v + 2*khalf.
    v2f Bm[6];
    #pragma unroll
    for (int m = 0; m < 6; ++m) {
        const int a0 = 4 * m + 2 * khalf;
        const int d0 = a0 - ln, d1 = a0 + 1 - ln;
        Bm[m][0] = (d0 >= 0 && d0 <= 6) ? (1.0f / 49.0f) : 0.0f;
        Bm[m][1] = (d1 >= 0 && d1 <= 6) ? (1.0f / 49.0f) : 0.0f;
    }

    float acc = 0.0f;
    for (int blk = wid; blk < 16; blk += 8) {       // 2 blocks per wave, none idle
        const int r0 = (blk >> 2) * 16;
        const int cb = (blk & 3) * 16;
        v8f D[5];
        #pragma unroll
        for (int q = 0; q < 5; ++q) {
            v8f cacc = {0.f, 0.f, 0.f, 0.f, 0.f, 0.f, 0.f, 0.f};
            // A 16x4 layout: lane holds row r0+ln, cols base+2*khalf+{0,1}
            const float* Vq = V + q * VQ + (r0 + ln) * VST + cb + 2 * khalf;
            #pragma unroll
            for (int m = 0; m < 6; ++m) {
                const v2f a = *reinterpret_cast<const v2f*>(Vq + 4 * m); // 8B-aligned
                cacc = __builtin_amdgcn_wmma_f32_16x16x4_f32(
                    false, a, false, Bm[m], (short)0, cacc, false, false);
            }
            D[q] = cacc;
        }
        const int cg = gx0 + cb + ln;               // D col N = lane&15
        #pragma unroll
        for (int v = 0; v < 8; ++v) {               // D row M = v + 8*khalf
            const float ux = D[0][v], uy = D[1][v];
            const float uxx = D[2][v], uyy = D[3][v], uxy = D[4][v];
            const float vx_ = CN * (uxx - ux * ux);
            const float vy_ = CN * (uyy - uy * uy);
            const float vxy = CN * (uxy - ux * uy);
            const float num = (2.0f * ux * uy + c1) * (2.0f * vxy + c2);
            const float den = (ux * ux + uy * uy + c1) * (vx_ + vy_ + c2);
            const float s = num * __builtin_amdgcn_rcpf(den);
            const int rg = gy0 + r0 + khalf * 8 + v;
            if (rg < WOUT && cg < WOUT) acc += s;   // ownership mask (select, not mul)
        }
    }

    // ---- block reduction (wave32-aware), deterministic partial per block ----
    #pragma unroll
    for (int off = 16; off > 0; off >>= 1) acc += __shfl_down(acc, off, 32);
    if (lane == 0) red[wid] = acc;
    __syncthreads();
    if (tid == 0) {
        float s = 0.0f;
        #pragma unroll
        for (int w = 0; w < 8; ++w) s += red[w];
        partial[b * (NT * NT) + tileId] = s;
    }
}

__global__ __launch_bounds__(NBLK)
void ssim_finalize(const float* __restrict__ partial, float* __restrict__ out, int n)
{
    __shared__ double sred[8];
    double a = 0.0;
    for (int i = threadIdx.x; i < n; i += NBLK) a += (double)partial[i];
    #pragma unroll
    for (int off = 16; off > 0; off >>= 1) a += __shfl_down(a, off, 32);
    const int lane = threadIdx.x & 31, wid = threadIdx.x >> 5;
    if (lane == 0) sred[wid] = a;
    __syncthreads();
    if (threadIdx.x == 0) {
        double t = 0.0;
        #pragma unroll
        for (int w = 0; w < 8; ++w) t += sred[w];
        const double cnt = 32.0 * (double)WOUT * (double)WOUT;  // 12,862,592
        out[0] = (float)(1.0 - t / cnt);
    }
}

extern "C" void kernel_launch(void* const* d_in, const int* in_sizes, int n_in,
                              void* d_out, int out_size, void* d_ws, size_t ws_size,
                              hipStream_t stream)
{
    (void)in_sizes; (void)n_in; (void)out_size; (void)ws_size;
    const float* x  = (const float*)d_in[0];
    const float* y  = (const float*)d_in[1];
    const float* dr = (const float*)d_in[2];
    float* out      = (float*)d_out;
    float* partial  = (float*)d_ws;                 // 3200 floats of scratch

    const size_t shmem = (size_t)LDS_FLTS * sizeof(float);   // ~126 KB (<320KB/WG)
    (void)hipFuncSetAttribute((const void*)ssim_tile_kernel,
                              hipFuncAttributeMaxDynamicSharedMemorySize, (int)shmem);

    dim3 grid(NT * NT, 32, 1);
    ssim_tile_kernel<<<grid, NBLK, shmem, stream>>>(x, y, dr, partial);
    ssim_finalize<<<1, NBLK, 0, stream>>>(partial, out, 32 * NT * NT);
}